// CannyEdgeDetector_77197742178948
// MI455X (gfx1250) — compile-verified
//
#include <hip/hip_runtime.h>

// Fused Canny pipeline for MI455X (gfx1250, wave32).
// One pass over HBM: tile+halo staged in LDS, vertical gaussian done on the
// matrix pipe via V_WMMA_F32_16X16X4_F32 (banded-matrix formulation).

typedef float v2f __attribute__((ext_vector_type(2)));
typedef float v8f __attribute__((ext_vector_type(8)));

#define TILE 28   // output tile (28x28)  -> blurred region 32x32 (2x2 WMMA tiles)
#define BLK  128  // 4 waves of 32

__global__ __launch_bounds__(BLK)
void canny_fused_kernel(const float* __restrict__ img,
                        const float* __restrict__ gauss, // 5 taps, unnormalized
                        float* __restrict__ out,
                        int B, int H, int W)
{
    // LDS staging (per block ~24 KB; 320 KB/WGP -> deep occupancy)
    __shared__ float gTab[32];       // gauss taps, zero-padded (branchless band lookup)
    __shared__ float sImg [36][37];  // input tile + halo4
    __shared__ float sHB  [36][33];  // horizontal blur (rows halo4, cols halo2)
    __shared__ float sBlur[32][33];  // separable blur  (halo2)
    __shared__ float sMag [30][31];  // channel-summed |grad| (halo1)
    __shared__ float sGx  [28][28];  // channel-summed gx at output pixels
    __shared__ float sGy  [28][28];

    const int tx   = threadIdx.x;
    const int lane = tx & 31;
    const int wave = tx >> 5;
    const int oy   = blockIdx.y * TILE;
    const int ox   = blockIdx.x * TILE;
    const int bz   = blockIdx.z;
    const long long HW = (long long)H * W;

    const float g0 = gauss[0], g1 = gauss[1], g2 = gauss[2],
                g3 = gauss[3], g4 = gauss[4];

    if (tx < 32) gTab[tx] = (tx < 5) ? gauss[tx] : 0.f;
    for (int t = tx; t < 30*31; t += BLK) ((float*)sMag)[t] = 0.f;
    for (int t = tx; t < 28*28; t += BLK) { ((float*)sGx)[t] = 0.f; ((float*)sGy)[t] = 0.f; }
    __syncthreads();

    // ---- WMMA A operand: banded gaussian matrix, G[m][k] = g[k-m] ----
    // A-layout (16x4 f32): lanes 0-15 = M rows; VGPR0 = K{0|2}, VGPR1 = K{1|3}.
    // Channel-invariant: build all 5 K-slices once, branchlessly via the
    // zero-padded LDS table (out-of-band offsets wrap into zeros).
    const int m  = lane & 15;
    const int kh = (lane >> 4) << 1;
    v2f aS[5];
    #pragma unroll
    for (int s = 0; s < 5; ++s) {
        int d0 = s*4 + kh - m;                 // band offset in [-15, 18]
        aS[s].x = gTab[d0 & 31];
        aS[s].y = gTab[(d0 + 1) & 31];
    }

    for (int ch = 0; ch < 3; ++ch) {
        const float* src = img + (long long)(bz*3 + ch) * HW;

        __syncthreads();  // previous iteration's LDS consumers are done

        // ---- stage input tile (zero beyond image = SAME padding of stage 1) ----
        for (int t = tx; t < 36*36; t += BLK) {
            int r = t / 36, c = t - r*36;
            int yy = oy - 4 + r, xx = ox - 4 + c;
            float v = 0.f;
            if ((unsigned)yy < (unsigned)H && (unsigned)xx < (unsigned)W) {
                long long off = (long long)yy * W + xx;
                v = src[off];
                if (ch < 2) __builtin_prefetch(src + HW + off, 0, 1); // next channel -> global_prefetch_b8
            }
            sImg[r][c] = v;
        }
        __syncthreads();

        // ---- horizontal gaussian (5 taps) ----
        for (int t = tx; t < 36*32; t += BLK) {
            int r = t >> 5, c = t & 31;
            sHB[r][c] = g0*sImg[r][c]   + g1*sImg[r][c+1] + g2*sImg[r][c+2]
                      + g3*sImg[r][c+3] + g4*sImg[r][c+4];
        }
        __syncthreads();

        // ---- vertical gaussian on the matrix pipe ----
        // blurred[32x32] = G[32x36] * hblur[36x32]; one 16x16 D tile per wave,
        // K band of 20 -> 5 x V_WMMA_F32_16X16X4_F32, B straight from LDS.
        {
            const int R = (wave >> 1) << 4;    // tile row base {0,16}
            const int N = (wave &  1) << 4;    // tile col base {0,16}
            v8f acc = {};
            #pragma unroll
            for (int s = 0; s < 5; ++s) {
                int br = R + s*4 + kh;         // B 4x16: row-striped like C/D
                v2f bb;
                bb.x = sHB[br    ][N + m];
                bb.y = sHB[br + 1][N + m];
                acc = __builtin_amdgcn_wmma_f32_16x16x4_f32(
                          false, aS[s], false, bb, (short)0, acc, false, false);
            }
            const int half = (lane >> 4) << 3; // C/D: VGPR v -> M = v (+8 for hi lanes)
            #pragma unroll
            for (int v = 0; v < 8; ++v) {
                int row = R + v + half, col = N + m;
                int yy = oy - 2 + row, xx = ox - 2 + col;
                bool in = ((unsigned)yy < (unsigned)H) && ((unsigned)xx < (unsigned)W);
                sBlur[row][col] = in ? acc[v] : 0.f;   // zero-pad stage-2 output
            }
        }
        __syncthreads();

        // ---- sobel + magnitude (accumulate over channels) ----
        for (int t = tx; t < 30*30; t += BLK) {
            int r = t / 30, c = t - r*30;
            float b00 = sBlur[r  ][c], b01 = sBlur[r  ][c+1], b02 = sBlur[r  ][c+2];
            float b10 = sBlur[r+1][c],                         b12 = sBlur[r+1][c+2];
            float b20 = sBlur[r+2][c], b21 = sBlur[r+2][c+1], b22 = sBlur[r+2][c+2];
            float gx = (b00 - b02) + 2.f*(b10 - b12) + (b20 - b22);
            float gy = (b00 + 2.f*b01 + b02) - (b20 + 2.f*b21 + b22);
            int yy = oy - 1 + r, xx = ox - 1 + c;
            bool in = ((unsigned)yy < (unsigned)H) && ((unsigned)xx < (unsigned)W);
            sMag[r][c] += in ? sqrtf(gx*gx + gy*gy) : 0.f;   // zero-pad stage-3 output
            if (r >= 1 && r <= 28 && c >= 1 && c <= 28) {
                sGx[r-1][c-1] += gx;
                sGy[r-1][c-1] += gy;
            }
        }

        // ---- emit blurred channel ----
        float* bOut = out + (long long)(bz*3 + ch) * HW;
        for (int t = tx; t < TILE*TILE; t += BLK) {
            int i = t / TILE, j = t - i*TILE;
            int yy = oy + i, xx = ox + j;
            if (yy < H && xx < W)
                bOut[(long long)yy * W + xx] = sBlur[i+2][j+2];
        }
    }
    __syncthreads();

    // ---- orientation, NMS, thresholds (5 remaining outputs) ----
    const long long O1 = (long long)B * 3 * HW;
    float* magOut   = out + O1 + (long long)bz * HW;
    float* oriOut   = magOut   + (long long)B * HW;
    float* thinOut  = oriOut   + (long long)B * HW;
    float* thrOut   = thinOut  + (long long)B * HW;
    float* earlyOut = thrOut   + (long long)B * HW;

    // dir filter k == mag[p] - mag[p + (dr,dc)]; (dr+1),(dc+1) packed 2b/entry
    const unsigned drBits = 0x01A9u;  // dr: 0,1,1,1,0,-1,-1,-1
    const unsigned dcBits = 0x901Au;  // dc: 1,1,0,-1,-1,-1,0,1

    for (int t = tx; t < TILE*TILE; t += BLK) {
        int i = t / TILE, j = t - i*TILE;
        int yy = oy + i, xx = ox + j;
        if (yy >= H || xx >= W) continue;
        float mg = sMag[i+1][j+1];
        float o  = atan2f(sGy[i][j], sGx[i][j]) * (180.0f / 3.14159f);
        float q  = rintf((o + 180.0f) / 45.0f);         // round-half-even, q in [0,8]
        int iq = (int)q;
        int ip = iq & 7, ineg = (iq + 4) & 7;
        int drp = (int)((drBits >> (2*ip  )) & 3u) - 1, dcp = (int)((dcBits >> (2*ip  )) & 3u) - 1;
        int drn = (int)((drBits >> (2*ineg)) & 3u) - 1, dcn = (int)((dcBits >> (2*ineg)) & 3u) - 1;
        float pos = mg - sMag[i+1+drp][j+1+dcp];
        float neg = mg - sMag[i+1+drn][j+1+dcn];
        float thin = (fminf(pos, neg) > 0.f) ? mg : 0.f;
        long long idx = (long long)yy * W + xx;
        magOut[idx]   = mg;
        oriOut[idx]   = q * 45.f;
        thinOut[idx]  = thin;
        thrOut[idx]   = (thin < 10.f) ? 0.f : thin;
        earlyOut[idx] = (mg < 10.f) ? 0.f : mg;
    }
}

extern "C" void kernel_launch(void* const* d_in, const int* in_sizes, int n_in,
                              void* d_out, int out_size, void* d_ws, size_t ws_size,
                              hipStream_t stream) {
    const float* img   = (const float*)d_in[0];
    const float* gauss = (const float*)d_in[1];
    float* out = (float*)d_out;
    const int H = 1024, W = 1024;
    const int B = in_sizes[0] / (3 * H * W);
    dim3 grid((W + TILE - 1) / TILE, (H + TILE - 1) / TILE, B);
    canny_fused_kernel<<<grid, BLK, 0, stream>>>(img, gauss, out, B, H, W);
}